// RankingTripletLoss_50397146251787
// MI455X (gfx1250) — compile-verified
//
#include <hip/hip_runtime.h>
#include <hip/hip_bf16.h>
#include <math.h>

typedef float v2f __attribute__((ext_vector_type(2)));
typedef float v4f __attribute__((ext_vector_type(4)));
typedef float v8f __attribute__((ext_vector_type(8)));

#define B_      1024
#define D_      2048
#define N_      64
#define NPAIRS  2016          // 64*63/2, divisible by 4
#define MARGIN_ 1.0f
#define EPS_    1e-8f

// ---------------------------------------------------------------------------
// Phase 1: bandwidth-bound streamer. One wave per (n,b) row: reduce 2048 f32.
// caption = 512 MB read-once  -> non-temporal loads (keep L2 for sentence)
// sentence = 8 MB, reused 64x -> regular-temporal loads (L2 resident)
// Outputs dot[n*B+b], cnorm2[n*B+b]; n==0 waves also write snorm2[b].
// ---------------------------------------------------------------------------
__global__ __launch_bounds__(256) void rank_phase1(
    const float* __restrict__ sent, const float* __restrict__ cap,
    float* __restrict__ dot_ws, float* __restrict__ cn2_ws,
    float* __restrict__ sn2_ws) {
  const int lane = threadIdx.x & 31;
  const int wave = threadIdx.x >> 5;
  const int row  = blockIdx.x * 8 + wave;          // row = n*B + b
  const int n = row >> 10;
  const int b = row & (B_ - 1);

  const v4f* __restrict__ crow = (const v4f*)(cap + ((size_t)n * B_ + b) * D_);
  const v4f* __restrict__ srow = (const v4f*)(sent + (size_t)b * D_);

  float dot = 0.f, cn2 = 0.f, sn2 = 0.f;
#pragma unroll
  for (int t = 0; t < D_ / (32 * 4); ++t) {        // 16 iters of b128/lane
    v4f c4 = __builtin_nontemporal_load(&crow[t * 32 + lane]);
    v4f s4 = srow[t * 32 + lane];
    dot += c4.x * s4.x + c4.y * s4.y + c4.z * s4.z + c4.w * s4.w;
    cn2 += c4.x * c4.x + c4.y * c4.y + c4.z * c4.z + c4.w * c4.w;
    sn2 += s4.x * s4.x + s4.y * s4.y + s4.z * s4.z + s4.w * s4.w;
  }
  // wave32 butterfly reduction
#pragma unroll
  for (int off = 16; off; off >>= 1) {
    dot += __shfl_xor(dot, off, 32);
    cn2 += __shfl_xor(cn2, off, 32);
    sn2 += __shfl_xor(sn2, off, 32);
  }
  if (lane == 0) {
    dot_ws[row] = dot;
    cn2_ws[row] = cn2;
    if (n == 0) sn2_ws[b] = sn2;
  }
}

// ---------------------------------------------------------------------------
// Phase 2: weighted pair-sum via V_WMMA_F32_16X16X4_F32.
// One wave per 16-wide b-tile. D = sum_k A*B + C with
//   A[m=b_off, k=pair] = relu(sim[j_k] - sim[i_k] + margin)
//   B[k, *]            = 1/(j_k - i_k)    (broadcast across all 16 columns)
// 504 WMMA steps consume all 2016 (i<j) pairs; column N=0 of the f32
// accumulator is per_batch[b0..b0+15].
// ISA A-layout (16x4 f32): lanes 0-15 -> K=0(v0),K=1(v1); lanes 16-31 -> K=2,K=3.
// ISA C/D layout: VGPR r: lanes0-15 = M=r, lanes16-31 = M=8+r  -> column 0
// lives in lane 0 (M=0..7) and lane 16 (M=8..15).
// ---------------------------------------------------------------------------
__global__ __launch_bounds__(32) void rank_phase2(
    const float* __restrict__ dot_ws, const float* __restrict__ cn2_ws,
    const float* __restrict__ sn2_ws, const int* __restrict__ order,
    float* __restrict__ pb) {
  __shared__ float simt[N_][16];
  __shared__ int   pi[NPAIRS];
  __shared__ int   pj[NPAIRS];
  __shared__ float pw[NPAIRS];

  const int lane = threadIdx.x;
  const int b0   = blockIdx.x * 16;

  // Build (i,j,weight) pair tables cooperatively (uniform loop, masked store).
  int p = 0;
  for (int i = 0; i < N_; ++i)
    for (int j = i + 1; j < N_; ++j) {
      if ((p & 31) == lane) {
        pi[p] = i; pj[p] = j; pw[p] = 1.0f / (float)(j - i);
      }
      ++p;
    }

  // Load ordered sim tile: simt[r][m] = cos-sim of caption order[r] vs batch b0+m.
  for (int t = 0; t < (N_ * 16) / 32; ++t) {
    int idx = t * 32 + lane;
    int r = idx >> 4, m = idx & 15;
    int src = order[r] - 1;                         // 1-based -> 0-based
    src = src < 0 ? 0 : (src > N_ - 1 ? N_ - 1 : src);
    int bm = b0 + m;
    float sn = sqrtf(sn2_ws[bm]);
    float cn = sqrtf(cn2_ws[(size_t)src * B_ + bm]);
    simt[r][m] = dot_ws[(size_t)src * B_ + bm] /
                 (fmaxf(sn, EPS_) * fmaxf(cn, EPS_));
  }
  __syncthreads();

  const int m   = lane & 15;           // A-matrix row M (b offset in tile)
  const int khi = (lane >> 4) << 1;    // 0 for lanes 0-15, 2 for lanes 16-31

  v8f acc = {};
  for (int t = 0; t < NPAIRS / 4; ++t) {
    int p0 = t * 4 + khi;
    int p1 = p0 + 1;
    float d0 = simt[pj[p0]][m] - simt[pi[p0]][m] + MARGIN_;
    float d1 = simt[pj[p1]][m] - simt[pi[p1]][m] + MARGIN_;
    v2f a, w;
    a.x = fmaxf(d0, 0.0f);
    a.y = fmaxf(d1, 0.0f);
    w.x = pw[p0];
    w.y = pw[p1];
    acc = __builtin_amdgcn_wmma_f32_16x16x4_f32(
        /*neg_a=*/false, a, /*neg_b=*/false, w,
        /*c_mod=*/(short)0, acc, /*reuse_a=*/false, /*reuse_b=*/false);
  }

  // Extract column N=0 of the accumulator: per_batch for this b-tile.
  if (lane == 0) {
#pragma unroll
    for (int r = 0; r < 8; ++r) pb[b0 + r] = acc[r];
  }
  if (lane == 16) {
#pragma unroll
    for (int r = 0; r < 8; ++r) pb[b0 + 8 + r] = acc[r];
  }
}

// ---------------------------------------------------------------------------
// Phase 3: deterministic final mean (no float atomics).
// loss = sum_b per_batch[b] / (NPAIRS * B)
// ---------------------------------------------------------------------------
__global__ __launch_bounds__(256) void rank_phase3(const float* __restrict__ pb,
                                                   float* __restrict__ out) {
  __shared__ float red[8];
  const int lane = threadIdx.x & 31;
  const int wave = threadIdx.x >> 5;
  float s = pb[threadIdx.x] + pb[threadIdx.x + 256] +
            pb[threadIdx.x + 512] + pb[threadIdx.x + 768];
#pragma unroll
  for (int off = 16; off; off >>= 1) s += __shfl_xor(s, off, 32);
  if (lane == 0) red[wave] = s;
  __syncthreads();
  if (threadIdx.x == 0) {
    float tot = 0.f;
#pragma unroll
    for (int w = 0; w < 8; ++w) tot += red[w];
    out[0] = tot / ((float)NPAIRS * (float)B_);
  }
}

// ---------------------------------------------------------------------------
extern "C" void kernel_launch(void* const* d_in, const int* in_sizes, int n_in,
                              void* d_out, int out_size, void* d_ws, size_t ws_size,
                              hipStream_t stream) {
  const float* sent  = (const float*)d_in[0];   // [B, D] f32
  const float* cap   = (const float*)d_in[1];   // [N, B, D] f32
  const int*   order = (const int*)d_in[2];     // [1, N] (1-based)
  float* out = (float*)d_out;

  // Workspace layout (floats): dot[N*B] | cn2[N*B] | sn2[B] | per_batch[B]
  float* dot_ws = (float*)d_ws;
  float* cn2_ws = dot_ws + (size_t)N_ * B_;
  float* sn2_ws = cn2_ws + (size_t)N_ * B_;
  float* pb_ws  = sn2_ws + B_;

  // Phase 1: 65536 rows, 1 wave each, 8 waves per block.
  rank_phase1<<<(N_ * B_) / 8, 256, 0, stream>>>(sent, cap, dot_ws, cn2_ws, sn2_ws);
  // Phase 2: 64 b-tiles of 16, one wave each (WMMA pair contraction).
  rank_phase2<<<B_ / 16, 32, 0, stream>>>(dot_ws, cn2_ws, sn2_ws, order, pb_ws);
  // Phase 3: final reduction -> scalar loss.
  rank_phase3<<<1, 256, 0, stream>>>(pb_ws, out);
}